// MCGRU_18365280158346
// MI455X (gfx1250) — compile-verified
//
#include <hip/hip_runtime.h>

// ---------------------------------------------------------------------------
// Problem constants (from the reference)
// ---------------------------------------------------------------------------
#define LAB  64
#define DEMO 16
#define HID  32
#define FEAT 8
#define BS   4096
#define TSEQ 128
#define CCOL (LAB * FEAT + HID)   // 544 = concat width

typedef float v2f __attribute__((ext_vector_type(2)));
typedef float v8f __attribute__((ext_vector_type(8)));

// ---------------------------------------------------------------------------
// Generic f32 WMMA GEMM:  C[M x N] = A[M x K] @ B[K x N] + bias[N]
// One wave (32 threads) per 16x16 output tile, K-loop of v_wmma_f32_16x16x4_f32.
// A: row-major with row stride strideA (lets us view x[:, -1, :] in place).
// transposeOut: store C[n*ldc + m] instead of C[m*ldc + n] (used to produce
// lastT = (LAB, BS) so each lab's GRU sequence is contiguous).
// ---------------------------------------------------------------------------
__global__ __launch_bounds__(32)
void mcgru_wmma_gemm_f32(const float* __restrict__ A, int strideA,
                         const float* __restrict__ B, int ldb,
                         const float* __restrict__ bias,
                         float* __restrict__ C, int ldc,
                         int tilesN, int K, int transposeOut)
{
    const int tile = blockIdx.x;
    const int tr   = tile / tilesN;
    const int tc   = tile % tilesN;
    const int row0 = tr << 4;
    const int col0 = tc << 4;

    const int lane = threadIdx.x;
    const int half = lane >> 4;    // 0: K pair {0,1} / M 0..7   1: K pair {2,3} / M 8..15
    const int mn   = lane & 15;

    v8f acc = {0.f, 0.f, 0.f, 0.f, 0.f, 0.f, 0.f, 0.f};

    const float* __restrict__ arow = A + (size_t)(row0 + mn) * (size_t)strideA;

    for (int k0 = 0; k0 < K; k0 += 4) {
        const int ka = k0 + (half << 1);
        // A 16x4: lane mn holds row M=mn; VGPR0/1 = K=ka, ka+1 (8B contiguous load)
        v2f a = *(const v2f*)(arow + ka);
        // B 4x16: lane mn holds col N=mn; VGPR0/1 = K=ka, ka+1
        v2f b;
        b.x = B[(size_t)ka       * (size_t)ldb + col0 + mn];
        b.y = B[(size_t)(ka + 1) * (size_t)ldb + col0 + mn];
        acc = __builtin_amdgcn_wmma_f32_16x16x4_f32(
                  /*neg_a=*/false, a, /*neg_b=*/false, b,
                  /*c_mod=*/(short)0, acc, /*reuse_a=*/false, /*reuse_b=*/false);
    }

    const float bv = bias[col0 + mn];

    if (!transposeOut) {
#pragma unroll
        for (int g = 0; g < 8; ++g) {
            const int m = row0 + g + (half << 3);
            C[(size_t)m * (size_t)ldc + col0 + mn] = acc[g] + bv;
        }
    } else {
#pragma unroll
        for (int g = 0; g < 8; ++g) {
            const int m = row0 + g + (half << 3);
            C[(size_t)(col0 + mn) * (size_t)ldc + m] = acc[g] + bv;
        }
    }
}

// ---------------------------------------------------------------------------
// GRU scan. 64 independent chains of 4096 sequential steps, hidden dim 8.
// Mapping: 8 lanes per lab (lane k owns hidden unit k and computes its r/z/n
// gates entirely locally -> only the h broadcast needs cross-lane shuffles),
// 4 labs per wave32, 16 blocks total. x values double-buffered as float4 so
// the serial dependence chain never stalls on a global load.
// Writes hidden states straight into the concat matrix Cmat[BS][544] at
// column 32 + lab*8 + k (demo occupies columns 0..31).
// ---------------------------------------------------------------------------
__device__ __forceinline__ float mcgru_sigmoid(float x) {
    return 1.0f / (1.0f + __expf(-x));        // v_exp_f32 + div
}
__device__ __forceinline__ float mcgru_tanh(float x) {
    const float e = __expf(-2.0f * x);
    return (1.0f - e) / (1.0f + e);
}

__global__ __launch_bounds__(32)
void mcgru_gru_scan(const float* __restrict__ lastT,   // [LAB][BS]
                    const float* __restrict__ W_ih,    // [LAB][24][1]
                    const float* __restrict__ W_hh,    // [LAB][24][8]
                    const float* __restrict__ b_ih,    // [LAB][24]
                    const float* __restrict__ b_hh,    // [LAB][24]
                    float* __restrict__ Cmat)          // [BS][CCOL]
{
    const int lane = threadIdx.x;
    const int g    = lane >> 3;                 // lab group within wave: 0..3
    const int k    = lane & 7;                  // hidden unit: 0..7
    const int lab  = (blockIdx.x << 2) + g;

    // Per-lane weights: rows k (r), 8+k (z), 16+k (n) of this lab's matrices.
    const float* __restrict__ wh = W_hh + (size_t)lab * 24 * 8;
    float whr[8], whz[8], whn[8];
#pragma unroll
    for (int j = 0; j < 8; ++j) {
        whr[j] = wh[(k)      * 8 + j];
        whz[j] = wh[(8 + k)  * 8 + j];
        whn[j] = wh[(16 + k) * 8 + j];
    }
    const float wir = W_ih[(size_t)lab * 24 + k];
    const float wiz = W_ih[(size_t)lab * 24 + 8 + k];
    const float win = W_ih[(size_t)lab * 24 + 16 + k];
    const float bir = b_ih[(size_t)lab * 24 + k];
    const float biz = b_ih[(size_t)lab * 24 + 8 + k];
    const float bin = b_ih[(size_t)lab * 24 + 16 + k];
    const float bhr = b_hh[(size_t)lab * 24 + k];
    const float bhz = b_hh[(size_t)lab * 24 + 8 + k];
    const float bhn = b_hh[(size_t)lab * 24 + 16 + k];

    const float* __restrict__ seq  = lastT + (size_t)lab * BS;
    float*       __restrict__ outp = Cmat + HID + lab * FEAT + k;

    float h = 0.0f;
    float4 buf = *(const float4*)seq;           // steps 0..3

    for (int t0 = 0; t0 < BS; t0 += 4) {
        float4 nextbuf;
        if (t0 + 4 < BS) {
            nextbuf = *(const float4*)(seq + t0 + 4);     // prefetch next block
        }
        if (t0 + 16 < BS) {
            __builtin_prefetch(seq + t0 + 16, 0, 0);      // global_prefetch_b8
        }
        const float xs[4] = {buf.x, buf.y, buf.z, buf.w};
#pragma unroll
        for (int u = 0; u < 4; ++u) {
            const float xt = xs[u];
            // Broadcast h across the 8 lanes of this lab group.
            float hj[8];
#pragma unroll
            for (int j = 0; j < 8; ++j) hj[j] = __shfl(h, (g << 3) + j, 32);
            float ghr = bhr, ghz = bhz, ghn = bhn;
#pragma unroll
            for (int j = 0; j < 8; ++j) {
                ghr = fmaf(whr[j], hj[j], ghr);
                ghz = fmaf(whz[j], hj[j], ghz);
                ghn = fmaf(whn[j], hj[j], ghn);
            }
            const float r = mcgru_sigmoid(fmaf(xt, wir, bir) + ghr);
            const float z = mcgru_sigmoid(fmaf(xt, wiz, biz) + ghz);
            const float n = mcgru_tanh(fmaf(xt, win, bin) + r * ghn);
            h = (1.0f - z) * n + z * h;
            outp[(size_t)(t0 + u) * CCOL] = h;
        }
        buf = nextbuf;
    }
}

// ---------------------------------------------------------------------------
// Launch: 4 kernels on one stream (stream order gives the dependencies).
//   1) lastT(64,4096)  = (x[:, -1, :] @ W_lab + b_lab)^T          [WMMA]
//   2) Cmat[:, 0:32]   = static @ W_demo + b_demo                 [WMMA]
//   3) Cmat[:, 32:544] = GRU scans over lastT                     [scan]
//   4) out             = Cmat @ W_out + b_out                     [WMMA]
// ---------------------------------------------------------------------------
extern "C" void kernel_launch(void* const* d_in, const int* in_sizes, int n_in,
                              void* d_out, int out_size, void* d_ws, size_t ws_size,
                              hipStream_t stream) {
    const float* x      = (const float*)d_in[0];   // (BS, T, LAB)
    const float* stat   = (const float*)d_in[1];   // (BS, DEMO)
    const float* W_demo = (const float*)d_in[2];   // (DEMO, HID)
    const float* b_demo = (const float*)d_in[3];   // (HID,)
    const float* W_lab  = (const float*)d_in[4];   // (LAB, LAB)
    const float* b_lab  = (const float*)d_in[5];   // (LAB,)
    const float* W_ih   = (const float*)d_in[6];   // (LAB, 24, 1)
    const float* W_hh   = (const float*)d_in[7];   // (LAB, 24, FEAT)
    const float* b_ih   = (const float*)d_in[8];   // (LAB, 24)
    const float* b_hh   = (const float*)d_in[9];   // (LAB, 24)
    const float* W_out  = (const float*)d_in[10];  // (CCOL, HID)
    const float* b_out  = (const float*)d_in[11];  // (HID,)
    float* out = (float*)d_out;                    // (BS, HID)

    float* lastT = (float*)d_ws;                           // LAB*BS floats (1 MB)
    float* Cmat  = lastT + (size_t)LAB * BS;               // BS*CCOL floats (~8.9 MB)

    // 1) lastT = (x_last @ W_lab + b_lab)^T ; x_last viewed in place via strides
    {
        const int tilesN = LAB / 16;                       // 4
        const int tiles  = (BS / 16) * tilesN;             // 1024
        mcgru_wmma_gemm_f32<<<tiles, 32, 0, stream>>>(
            x + (size_t)(TSEQ - 1) * LAB, TSEQ * LAB,      // A, strideA = 8192
            W_lab, LAB, b_lab,
            lastT, BS, tilesN, LAB, /*transposeOut=*/1);
    }

    // 2) Cmat[:, 0:32] = static @ W_demo + b_demo
    {
        const int tilesN = HID / 16;                       // 2
        const int tiles  = (BS / 16) * tilesN;             // 512
        mcgru_wmma_gemm_f32<<<tiles, 32, 0, stream>>>(
            stat, DEMO, W_demo, HID, b_demo,
            Cmat, CCOL, tilesN, DEMO, /*transposeOut=*/0);
    }

    // 3) GRU scans -> Cmat[:, 32:544]
    {
        mcgru_gru_scan<<<LAB / 4, 32, 0, stream>>>(
            lastT, W_ih, W_hh, b_ih, b_hh, Cmat);
    }

    // 4) out = Cmat @ W_out + b_out
    {
        const int tilesN = HID / 16;                       // 2
        const int tiles  = (BS / 16) * tilesN;             // 512
        mcgru_wmma_gemm_f32<<<tiles, 32, 0, stream>>>(
            Cmat, CCOL, W_out, HID, b_out,
            out, HID, tilesN, CCOL, /*transposeOut=*/0);
    }
}